// knrm_38457137168915
// MI455X (gfx1250) — compile-verified
//
#include <hip/hip_runtime.h>
#include <math.h>

#define VOCAB  50000
#define D_EMB  300
#define N_BINS 11
#define BB     64
#define LQ     32
#define LD     1024

#define NCHUNK 64      // doc rows per workgroup
#define KC     152     // K-chunk staged in LDS (divisible by 4; 2 chunks cover 300 w/ zero pad)
#define SLOTS  (KC/4)  // float4 slots per row per chunk

typedef __attribute__((ext_vector_type(2))) float v2f;
typedef __attribute__((ext_vector_type(8))) float v8f;

#if defined(__HIP_DEVICE_COMPILE__) && __has_builtin(__builtin_amdgcn_global_load_async_to_lds_b128)
#define USE_ASYNC_LDS 1
#else
#define USE_ASYNC_LDS 0
#endif

#if USE_ASYNC_LDS
// param1 type per compiler diagnostic: generic pointer to int __attribute__((vector_size(16)))
typedef int v4i __attribute__((vector_size(16)));
typedef __attribute__((address_space(3))) v4i* v4i_lds_ptr;

static __device__ __forceinline__ void async_copy_b128(const float* gsrc, float* lds_dst) {
    // Low 32 bits of a flat shared-memory address are the LDS byte offset
    // (ISA 10.2); rebuild an AS(3) pointer via inttoptr.
    unsigned lds_off = (unsigned)(unsigned long long)(size_t)lds_dst;
    __builtin_amdgcn_global_load_async_to_lds_b128(
        (v4i*)(void*)gsrc,
        (v4i_lds_ptr)(unsigned long long)lds_off,
        0, 0);
}

static __device__ __forceinline__ void wait_async_zero() {
#if __has_builtin(__builtin_amdgcn_s_wait_asynccnt)
    __builtin_amdgcn_s_wait_asynccnt(0);
#else
    asm volatile("s_wait_asynccnt 0x0" ::: "memory");
#endif
}
#endif

__constant__ float c_mu[N_BINS]   = {1.0f,0.9f,0.7f,0.5f,0.3f,0.1f,-0.1f,-0.3f,-0.5f,-0.7f,-0.9f};
// 1/(2*sigma^2): sigma0=1e-3 -> 5e5 ; others sigma=0.1 -> 50
__constant__ float c_hinv[N_BINS] = {5.0e5f,50.f,50.f,50.f,50.f,50.f,50.f,50.f,50.f,50.f,50.f};

// ---------------------------------------------------------------------------
// Kernel 1: inverse L2 norms of gathered embedding rows (1 wave per row)
// ---------------------------------------------------------------------------
__global__ __launch_bounds__(256)
void norms_kernel(const int* __restrict__ q_tok, const int* __restrict__ d_tok,
                  const float* __restrict__ emb,
                  float* __restrict__ inv_nq, float* __restrict__ inv_nd) {
    const int NROWS = BB * LQ + BB * LD;
    int wave = (blockIdx.x * blockDim.x + threadIdx.x) >> 5;
    int lane = threadIdx.x & 31;
    if (wave >= NROWS) return;
    int tok = (wave < BB * LQ) ? q_tok[wave] : d_tok[wave - BB * LQ];
    const float* row = emb + (size_t)tok * D_EMB;
    float s = 0.f;
    for (int i = lane; i < D_EMB; i += 32) { float x = row[i]; s += x * x; }
    #pragma unroll
    for (int off = 16; off > 0; off >>= 1) s += __shfl_xor(s, off, 32);
    if (lane == 0) {
        float inv = 1.0f / fmaxf(sqrtf(s), 1e-12f);
        if (wave < BB * LQ) inv_nq[wave] = inv;
        else                inv_nd[wave - BB * LQ] = inv;
    }
}

// ---------------------------------------------------------------------------
// Kernel 2: fused sim-GEMM (f32 WMMA 16x16x4) + RBF soft histogram pooling
// grid = (LD/NCHUNK, BB), block = 256 (8 waves, one 16x16 tile each)
// ---------------------------------------------------------------------------
__global__ __launch_bounds__(256)
void sim_rbf_kernel(const int* __restrict__ q_tok_all, const int* __restrict__ d_tok_all,
                    const float* __restrict__ emb,
                    const float* __restrict__ mask_d,
                    const float* __restrict__ inv_nq, const float* __restrict__ inv_nd,
                    float* __restrict__ pool) {
    __shared__ float q_lds[LQ * KC];          // 19456 B
    __shared__ float d_lds[NCHUNK * KC];      // 38912 B
    __shared__ float pool_lds[LQ * N_BINS];   //  1408 B
    __shared__ float qn[LQ];
    __shared__ float dn[NCHUNK];
    __shared__ float md[NCHUNK];
    __shared__ int   qtk[LQ];
    __shared__ int   dtk[NCHUNK];

    const int b     = blockIdx.y;
    const int chunk = blockIdx.x;
    const int tid   = threadIdx.x;

    for (int i = tid; i < LQ * N_BINS; i += 256) pool_lds[i] = 0.f;
    if (tid < LQ) {
        qn[tid]  = inv_nq[b * LQ + tid];
        qtk[tid] = q_tok_all[b * LQ + tid];
    }
    if (tid >= 64 && tid < 64 + NCHUNK) {
        int r  = tid - 64;
        int gd = b * LD + chunk * NCHUNK + r;
        dn[r]  = inv_nd[gd];
        md[r]  = mask_d[gd];
        dtk[r] = d_tok_all[gd];
    }

    const int w  = tid >> 5;           // wave id 0..7
    const int l  = tid & 31;           // lane
    const int mt = w & 1;              // M tile (0..1)
    const int nt = w >> 1;             // N tile (0..3)
    const int khalf = (l >> 4) * 2;    // lanes 16-31 hold K+2,K+3
    const float* qbase = &q_lds[(mt * 16 + (l & 15)) * KC];
    const float* dbase = &d_lds[(nt * 16 + (l & 15)) * KC];

    v8f acc = {};
    for (int kc = 0; kc < 2; kc++) {
        const int k0 = kc * KC;
        __syncthreads();   // previous chunk fully consumed; token/scale tables ready
#if USE_ASYNC_LDS
        // CDNA5 async gather: memory -> LDS directly (B128), tracked by ASYNCcnt
        for (int u = tid; u < LQ * SLOTS; u += 256) {
            int r = u / SLOTS, s4 = u % SLOTS;
            int col = k0 + s4 * 4;
            float* dst = &q_lds[r * KC + s4 * 4];
            if (col < D_EMB) {
                async_copy_b128(emb + (size_t)qtk[r] * D_EMB + col, dst);
            } else {
                *(float4*)dst = make_float4(0.f, 0.f, 0.f, 0.f);
            }
        }
        for (int u = tid; u < NCHUNK * SLOTS; u += 256) {
            int r = u / SLOTS, s4 = u % SLOTS;
            int col = k0 + s4 * 4;
            float* dst = &d_lds[r * KC + s4 * 4];
            if (col < D_EMB) {
                async_copy_b128(emb + (size_t)dtk[r] * D_EMB + col, dst);
            } else {
                *(float4*)dst = make_float4(0.f, 0.f, 0.f, 0.f);
            }
        }
        wait_async_zero();
#else
        // Vectorized B128 fallback: clamp + select (branch-free zero tail)
        for (int u = tid; u < LQ * SLOTS; u += 256) {
            int r = u / SLOTS, s4 = u % SLOTS;
            int col  = k0 + s4 * 4;
            int colc = (col < D_EMB) ? col : (D_EMB - 4);
            float4 v = *(const float4*)(emb + (size_t)qtk[r] * D_EMB + colc);
            if (col >= D_EMB) v = make_float4(0.f, 0.f, 0.f, 0.f);
            *(float4*)&q_lds[r * KC + s4 * 4] = v;
        }
        for (int u = tid; u < NCHUNK * SLOTS; u += 256) {
            int r = u / SLOTS, s4 = u % SLOTS;
            int col  = k0 + s4 * 4;
            int colc = (col < D_EMB) ? col : (D_EMB - 4);
            float4 v = *(const float4*)(emb + (size_t)dtk[r] * D_EMB + colc);
            if (col >= D_EMB) v = make_float4(0.f, 0.f, 0.f, 0.f);
            *(float4*)&d_lds[r * KC + s4 * 4] = v;
        }
#endif
        __syncthreads();
        #pragma unroll 4
        for (int k = 0; k < KC; k += 4) {
            v2f a  = *(const v2f*)(qbase + k + khalf);
            v2f bv = *(const v2f*)(dbase + k + khalf);
            // D = A(16x4 f32) x B(4x16 f32) + C, full f32 precision
            acc = __builtin_amdgcn_wmma_f32_16x16x4_f32(
                false, a, false, bv, (short)0, acc, false, false);
        }
    }

    // acc[v]: M = mt*16 + (l>>4)*8 + v ; N = nt*16 + (l&15)
    const int nloc  = nt * 16 + (l & 15);
    const int mbase = mt * 16 + (l >> 4) * 8;
    const float invd = dn[nloc];
    const float mk   = md[nloc];
    #pragma unroll
    for (int v = 0; v < 8; v++) {
        const int m = mbase + v;
        float sim = acc[v] * qn[m] * invd;
        #pragma unroll
        for (int kb = 0; kb < N_BINS; kb++) {
            float dlt = sim - c_mu[kb];
            float e = __expf(-dlt * dlt * c_hinv[kb]) * mk;
            atomicAdd(&pool_lds[m * N_BINS + kb], e);   // ds_add_f32
        }
    }
    __syncthreads();
    for (int i = tid; i < LQ * N_BINS; i += 256)
        atomicAdd(&pool[b * LQ * N_BINS + i], pool_lds[i]);  // global_atomic_add_f32
}

// ---------------------------------------------------------------------------
// Kernel 3: log / idf attention / dense / tanh  (1 wave per batch)
// ---------------------------------------------------------------------------
__global__ __launch_bounds__(32)
void finalize_kernel(const int* __restrict__ q_tok_all, const float* __restrict__ mask_q,
                     const float* __restrict__ attn_table,
                     const float* __restrict__ idf_w, const float* __restrict__ idf_b,
                     const float* __restrict__ dense_w, const float* __restrict__ dense_b,
                     const float* __restrict__ pool, float* __restrict__ out) {
    const int b = blockIdx.x;
    const int l = threadIdx.x;     // 0..31 == query position
    int tok = q_tok_all[b * LQ + l];
    float wq = mask_q[b * LQ + l] * (attn_table[tok] * idf_w[0] + idf_b[0]) * 0.01f;
    float s[N_BINS];
    #pragma unroll
    for (int k = 0; k < N_BINS; k++) {
        float p = pool[(b * LQ + l) * N_BINS + k];
        s[k] = logf(fmaxf(p, 1e-10f)) * wq;
    }
    #pragma unroll
    for (int k = 0; k < N_BINS; k++)
        #pragma unroll
        for (int off = 16; off > 0; off >>= 1)
            s[k] += __shfl_xor(s[k], off, 32);
    if (l == 0) {
        float acc = dense_b[0];
        #pragma unroll
        for (int k = 0; k < N_BINS; k++) acc += s[k] * dense_w[k];
        out[b] = tanhf(acc);
    }
}

// ---------------------------------------------------------------------------
extern "C" void kernel_launch(void* const* d_in, const int* in_sizes, int n_in,
                              void* d_out, int out_size, void* d_ws, size_t ws_size,
                              hipStream_t stream) {
    (void)in_sizes; (void)n_in; (void)out_size; (void)ws_size;
    const int*   inputs_q   = (const int*)d_in[0];
    const int*   inputs_d   = (const int*)d_in[1];
    const float* mask_q     = (const float*)d_in[2];
    const float* mask_d     = (const float*)d_in[3];
    const float* word_emb   = (const float*)d_in[4];
    const float* attn_table = (const float*)d_in[5];
    const float* idf_w      = (const float*)d_in[6];
    const float* idf_b      = (const float*)d_in[7];
    const float* dense_w    = (const float*)d_in[8];
    const float* dense_b    = (const float*)d_in[9];
    float* out = (float*)d_out;

    // workspace layout
    float* pool   = (float*)d_ws;                 // BB*LQ*N_BINS
    float* inv_nq = pool + BB * LQ * N_BINS;      // BB*LQ
    float* inv_nd = inv_nq + BB * LQ;             // BB*LD

    (void)hipMemsetAsync(pool, 0, (size_t)BB * LQ * N_BINS * sizeof(float), stream);

    {
        const int nrows = BB * LQ + BB * LD;      // 67584, 8 waves/block
        norms_kernel<<<(nrows + 7) / 8, 256, 0, stream>>>(
            inputs_q, inputs_d, word_emb, inv_nq, inv_nd);
    }
    {
        dim3 grid(LD / NCHUNK, BB);               // 16 x 64
        sim_rbf_kernel<<<grid, 256, 0, stream>>>(
            inputs_q, inputs_d, word_emb, mask_d, inv_nq, inv_nd, pool);
    }
    finalize_kernel<<<BB, 32, 0, stream>>>(
        inputs_q, mask_q, attn_table, idf_w, idf_b, dense_w, dense_b, pool, out);
}